// SimpleDense_91079076479203
// MI455X (gfx1250) — compile-verified
//
#include <hip/hip_runtime.h>
#include <stdint.h>

// Reference shapes: T=11, M=10, N_STATES=45, N=32768 samples.
// out[n, r, c] (45x45 per sample):
//   r<40, i=r>>2:
//     c==4j+7            -> inputs[n,j] * wout[i,j]   (res1, cols ==3 mod 4)
//     c in {4i..4i+2}    -> wout[i, n%10]             (res2, cols !=3 mod 4)
//   else 0.  (disjoint sets; rows 40..44 all zero; mask is identity)
//
// Store-bandwidth bound: 265 MB out @ 23.3 TB/s ~= 11.4 us floor.
// Strategy: build 4-sample tile (8100 f32 = 32400 B, 16B aligned) in LDS,
// stream it out with ONE Tensor-Data-Mover store per block (CDNA5 TDM path).

#define TM 10
#define NS 45
#define GROUP 4
#define TILE_F (GROUP * NS * NS)   // 8100 floats per block

typedef uint32_t __attribute__((ext_vector_type(4))) u32x4_t;
typedef int32_t  __attribute__((ext_vector_type(8))) i32x8_t;
typedef int32_t  __attribute__((ext_vector_type(4))) i32x4_t;

__global__ __launch_bounds__(256)
void simple_dense_tdm_kernel(const float* __restrict__ inputs,
                             const float* __restrict__ w,
                             float* __restrict__ out)
{
    __shared__ __attribute__((aligned(16))) float tile[TILE_F];
    __shared__ float wout_s[100];      // 10x10 expanded upper-triangular
    __shared__ float inp_s[GROUP][10]; // 4 samples' inputs

    const int tid = threadIdx.x;
    const int n0  = blockIdx.x * GROUP;

    // Phase 1: expand w -> wout (triu row-major pack), stage inputs, zero tile.
    if (tid < 100) {
        int i = tid / 10, j = tid % 10;
        float v = 0.0f;
        if (j >= i) v = w[i * TM - (i * (i - 1)) / 2 + (j - i)];
        wout_s[tid] = v;
    } else if (tid >= 128 && tid < 128 + GROUP * 10) {
        int t = tid - 128;
        int g = t / 10, j = t % 10;
        inp_s[g][j] = inputs[(size_t)(n0 + g) * TM + j];
    }
    const float4 z4 = make_float4(0.f, 0.f, 0.f, 0.f);
#pragma unroll
    for (int k = 0; k < 8; ++k) {
        int idx = tid + 256 * k;
        if (idx < TILE_F / 4) ((float4*)tile)[idx] = z4;   // ds_store_b128
    }
    __syncthreads();

    // Phase 2: scatter structurally-nonzero entries (disjoint addresses).
    // res1: 4 samples x 10 i x 10 j x 4 row-replicas = 1600 writes
    for (int t = tid; t < GROUP * 10 * 10 * 4; t += 256) {
        int g   = t / 400;
        int rem = t % 400;
        int i   = rem / 40;  rem %= 40;
        int j   = rem / 4;
        int a   = rem % 4;
        tile[g * (NS * NS) + (4 * i + a) * NS + (4 * j + 7)] =
            inp_s[g][j] * wout_s[i * 10 + j];
    }
    // res2: 4 samples x 10 i x 4 rows x 3 cols = 480 writes
    for (int t = tid; t < GROUP * 10 * 4 * 3; t += 256) {
        int g   = t / 120;
        int rem = t % 120;
        int i   = rem / 12;  rem %= 12;
        int a   = rem / 3;
        int b   = rem % 3;
        tile[g * (NS * NS) + (4 * i + a) * NS + (4 * i + b)] =
            wout_s[i * 10 + (n0 + g) % 10];
    }
    __syncthreads();

    // Phase 3: one TDM store per block: LDS tile -> global (32,400 B).
    // Issued by wave 0 only (instruction executes once per wave; EXEC ignored
    // by tensor ops). S_ENDPGM implicitly waits all counters, but we emit
    // s_wait_tensorcnt 0 explicitly.
    if (tid < 32) {
        uint64_t gaddr = (uint64_t)(uintptr_t)(out + (size_t)n0 * (NS * NS));
        uint32_t laddr = (uint32_t)(uintptr_t)&tile[0]; // LDS aperture: low 32 bits = LDS offset

        // D# group 0 (128b): count=1, lds_addr, global_addr[56:0], type=2
        u32x4_t g0;
        g0[0] = 1u;                       // [1:0] count=1 (valid), rest 0
        g0[1] = laddr;                    // [63:32] lds_addr (bytes)
        g0[2] = (uint32_t)gaddr;          // [95:64] global_addr lo
        g0[3] = (uint32_t)((gaddr >> 32) & 0x01FFFFFFu) | (2u << 30); // [120:96] + type=2

        // D# group 1 (256b): 1-row tensor of 8100 4-byte elements.
        i32x8_t g1;
        g1[0] = (int)(2u << 16);          // [17:16] data_size=2 (4B); mask=0; no pad/iter
        g1[1] = (int)(8100u << 16);       // [63:48]  tensor_dim0 lo16 (=8100)
        g1[2] = (int)(1u << 16);          // [95:80]  tensor_dim1 lo16 (=1)
        g1[3] = (int)(8100u << 16);       // [127:112] tile_dim0 = 8100
        g1[4] = 1;                        // [143:128] tile_dim1 = 1; tile_dim2 = 0
        g1[5] = 8100;                     // [191:160] tensor_dim0_stride lo32
        g1[6] = (int)(8100u << 16);       // [223:208] tensor_dim1_stride lo16
        g1[7] = 0;                        // [255:224] tensor_dim1_stride hi

        i32x4_t g2 = (i32x4_t)0;          // dims 2/3 unused
        i32x4_t g3 = (i32x4_t)0;
        i32x8_t g4 = (i32x8_t)0;          // 5th descriptor operand (uncharacterized): zero-fill

        __builtin_amdgcn_tensor_store_from_lds(g0, g1, g2, g3, g4, 0);
        __builtin_amdgcn_s_wait_tensorcnt(0);
    }
}

extern "C" void kernel_launch(void* const* d_in, const int* in_sizes, int n_in,
                              void* d_out, int out_size, void* d_ws, size_t ws_size,
                              hipStream_t stream) {
    (void)n_in; (void)d_ws; (void)ws_size; (void)out_size;
    const float* inputs = (const float*)d_in[0];   // (N, 10) f32
    const float* w      = (const float*)d_in[1];   // (55,)  f32
    float*       out    = (float*)d_out;           // (N*45*45,) f32

    const int N = in_sizes[0] / TM;                // 32768 (divisible by GROUP)
    const int blocks = N / GROUP;                  // 8192
    simple_dense_tdm_kernel<<<blocks, 256, 0, stream>>>(inputs, w, out);
}